// LearnableCLAHE_82042465288881
// MI455X (gfx1250) — compile-verified
//
#include <hip/hip_runtime.h>
#include <stdint.h>

// LearnableCLAHE for MI455X (gfx1250, wave32).
// x: [8,16,640,640] f32. grid 8x8 -> 80x80 cells.
// Roofline: 3 reads + 1 write of 210MB ~= 0.84GB -> ~36us @ 23.3 TB/s.
// Passes 1/2: b128 streaming reads (320-thread blocks, exactly 5 float4 per
// thread per 80x80 cell), WMMA-assisted block reduction.
// Pass 3: async global->LDS B128 2-deep pipeline (ASYNCcnt), b128 stores.

#define B_ 8
#define C_ 16
#define H_ 640
#define W_ 640
#define CELL 80
#define PLANE (H_*W_)      // 409600
#define NPLANE (B_*C_)     // 128

typedef float v2f __attribute__((ext_vector_type(2)));
typedef float v8f __attribute__((ext_vector_type(8)));

__device__ __forceinline__ float clip10(float v) {
  return fminf(fmaxf(v, -10.f), 10.f);
}

// Block-wide (320 threads = 10 full wave32) sum of per-thread partials.
// LDS fold 320->32 per-lane columns (redundantly in every wave so EXEC stays
// all-ones), then one V_WMMA_F32_16X16X4_F32 folds 32->16 (A = ones 16x4,
// B[0,n]=col[n] lanes0-15, B[2,n]=col[n+16] lanes16-31, B[1]=B[3]=0 so
// D[m,n] = col[n]+col[n+16]), then 4 wave shuffles finish 16->1.
__device__ __forceinline__ float block_sum_320(float partial, float* lds) {
  const int tid = threadIdx.x;
  lds[tid] = partial;
  __syncthreads();
  const int lane = tid & 31;
  float col = 0.f;
  #pragma unroll
  for (int i = 0; i < 10; ++i) col += lds[lane + 32*i];
  v2f a; a.x = 1.f; a.y = 1.f;       // A = ones(16x4)
  v2f b; b.x = col; b.y = 0.f;       // B rows 0,2 = lane partials
  v8f c = {0.f,0.f,0.f,0.f,0.f,0.f,0.f,0.f};
  v8f d = __builtin_amdgcn_wmma_f32_16x16x4_f32(false, a, false, b,
                                                (short)0, c, false, false);
  float v = d[0];                    // lanes 0..15: col[n]+col[n+16]
  v += __shfl_down(v, 8);
  v += __shfl_down(v, 4);
  v += __shfl_down(v, 2);
  v += __shfl_down(v, 1);
  return __shfl(v, 0);               // identical in every wave
}

// ---------------- pass 1: per-cell mean of clipped x ----------------
__global__ __launch_bounds__(320) void cell_mean_kernel(
    const float* __restrict__ x, float* __restrict__ mean) {
  __shared__ float lds[320];
  const int bid = blockIdx.x;        // [0, 8192)
  const int plane = bid >> 6;
  const int cellid = bid & 63;
  const int gi = cellid >> 3, gj = cellid & 7;
  const float* base = x + (size_t)plane * PLANE + gi*CELL*W_ + gj*CELL;
  float partial = 0.f;
  #pragma unroll
  for (int i = 0; i < 5; ++i) {      // 1600 float4 / 320 threads
    int idx4 = i*320 + threadIdx.x;
    int r = idx4 / 20, c4 = idx4 - r*20;
    const float4 xv = *(const float4*)(base + r*W_ + c4*4);
    partial += clip10(xv.x) + clip10(xv.y) + clip10(xv.z) + clip10(xv.w);
  }
  float tot = block_sum_320(partial, lds);
  if (threadIdx.x == 0) mean[bid] = tot * (1.f/6400.f);
}

// ---------------- pass 2: per-cell std of (x - bilinear(mean)) ----------------
__global__ __launch_bounds__(320) void cell_std_kernel(
    const float* __restrict__ x, const float* __restrict__ mean,
    float* __restrict__ stdo) {
  __shared__ float lds[320];
  __shared__ float nb[9];            // 3x3 clamped mean neighborhood
  const int bid = blockIdx.x;
  const int plane = bid >> 6;
  const int cellid = bid & 63;
  const int gi = cellid >> 3, gj = cellid & 7;
  if (threadIdx.x < 9) {
    int di = threadIdx.x / 3 - 1, dj = threadIdx.x % 3 - 1;
    int ci = min(max(gi + di, 0), 7), cj = min(max(gj + dj, 0), 7);
    nb[threadIdx.x] = mean[plane*64 + ci*8 + cj];
  }
  __syncthreads();
  const float* base = x + (size_t)plane * PLANE + gi*CELL*W_ + gj*CELL;
  float partial = 0.f;
  for (int i = 0; i < 5; ++i) {
    int idx4 = i*320 + threadIdx.x;
    int r = idx4 / 20, c4 = idx4 - r*20;
    const float4 xv = *(const float4*)(base + r*W_ + c4*4);
    const float xe[4] = {xv.x, xv.y, xv.z, xv.w};
    // row-constant h interp for this float4
    int h = gi*CELL + r;
    float sh = fmaxf((h + 0.5f)*0.0125f - 0.5f, 0.f);
    int h0 = min((int)sh, 7);
    float fh = sh - (float)h0;
    int h1 = min(h0 + 1, 7);
    int lh0 = h0 - gi + 1, lh1 = h1 - gi + 1;    // in {0,1,2}
    #pragma unroll
    for (int e = 0; e < 4; ++e) {
      int w = gj*CELL + c4*4 + e;
      float sw = fmaxf((w + 0.5f)*0.0125f - 0.5f, 0.f);
      int w0 = min((int)sw, 7);
      float fw = sw - (float)w0;
      int w1 = min(w0 + 1, 7);
      int lw0 = w0 - gj + 1, lw1 = w1 - gj + 1;
      float m00 = nb[lh0*3+lw0], m01 = nb[lh0*3+lw1];
      float m10 = nb[lh1*3+lw0], m11 = nb[lh1*3+lw1];
      float mu = (m00*(1.f-fw) + m01*fw)*(1.f-fh)
               + (m10*(1.f-fw) + m11*fw)*fh;
      float dlt = clip10(xe[e]) - mu;
      partial += dlt * dlt;
    }
  }
  float tot = block_sum_320(partial, lds);
  if (threadIdx.x == 0) stdo[bid] = sqrtf(tot*(1.f/6400.f) + 1e-8f);
}

// ---------------- pass 3: streamed elementwise apply ----------------
// One 160-thread block per 4 rows of one (b,c) plane. Rows are async-copied
// global->LDS (B128, one 16B chunk per lane) with 2-deep pipelining on
// ASYNCcnt; per-row h-lerped 8-entry mean/std tables live in LDS so the
// per-pixel work is a single w-lerp each.
__global__ __launch_bounds__(160) void apply_kernel(
    const float* __restrict__ x, const float* __restrict__ mean,
    const float* __restrict__ stdt, const float* __restrict__ alpha,
    const float* __restrict__ beta, float* __restrict__ out) {
  __shared__ float xbuf[2][W_];      // double-buffered row, 2*2560B
  __shared__ float rowm[4][8];
  __shared__ float rows[4][8];
  const int tid = threadIdx.x;
  const int bid = blockIdx.x;        // [0, 128*160)
  const int plane = bid / 160;
  const int rgrp  = bid - plane*160;
  const int hbase = rgrp * 4;
  const int cch = plane & 15;

  if (tid < 32) {                    // 4 rows x 8 cells: h-interp tables
    int ri = tid >> 3, cell = tid & 7;
    int h = hbase + ri;
    float sh = fmaxf((h + 0.5f)*0.0125f - 0.5f, 0.f);
    int h0 = min((int)sh, 7);
    float fh = sh - (float)h0;
    int h1 = min(h0 + 1, 7);
    const float* mp = mean + plane*64;
    const float* sp = stdt + plane*64;
    rowm[ri][cell] = mp[h0*8+cell]*(1.f-fh) + mp[h1*8+cell]*fh;
    rows[ri][cell] = sp[h0*8+cell]*(1.f-fh) + sp[h1*8+cell]*fh;
  }
  const float a  = fminf(fmaxf(alpha[cch], 0.1f), 2.0f);
  const float bb = fminf(fmaxf(beta[cch], -1.0f), 1.0f);

  // per-thread column lerp params (columns are fixed across the 4 rows)
  int w0e[4], w1e[4]; float fwe[4];
  #pragma unroll
  for (int e = 0; e < 4; ++e) {
    int w = tid*4 + e;
    float sw = fmaxf((w + 0.5f)*0.0125f - 0.5f, 0.f);
    int w0 = min((int)sw, 7);
    fwe[e] = sw - (float)w0;
    w0e[e] = w0;
    w1e[e] = min(w0 + 1, 7);
  }

  const size_t pbase = (size_t)plane * PLANE + (size_t)hbase * W_;
  const float* xrow = x + pbase;
  float* orow = out + pbase;

  // prefetch row 0 (each lane async-loads its own 16B chunk into LDS)
  {
    uint32_t lo = (uint32_t)(uintptr_t)(&xbuf[0][tid*4]);
    uint64_t ga = (uint64_t)(uintptr_t)(xrow + tid*4);
    asm volatile("global_load_async_to_lds_b128 %0, %1, off"
                 :: "v"(lo), "v"(ga) : "memory");
  }
  __syncthreads();                   // rowm/rows visible to all waves

  #pragma unroll
  for (int r = 0; r < 4; ++r) {
    if (r < 3) {
      uint32_t lo = (uint32_t)(uintptr_t)(&xbuf[(r+1)&1][tid*4]);
      uint64_t ga = (uint64_t)(uintptr_t)(xrow + (size_t)(r+1)*W_ + tid*4);
      asm volatile("global_load_async_to_lds_b128 %0, %1, off"
                   :: "v"(lo), "v"(ga) : "memory");
      asm volatile("s_wait_asynccnt 1" ::: "memory"); // row r done (in-order)
    } else {
      asm volatile("s_wait_asynccnt 0" ::: "memory");
    }
    const float4 xv = *(const float4*)(&xbuf[r & 1][tid*4]);
    const float xe[4] = {xv.x, xv.y, xv.z, xv.w};
    float res[4];
    #pragma unroll
    for (int e = 0; e < 4; ++e) {
      float xc = clip10(xe[e]);
      float omf = 1.f - fwe[e];
      float mu = rowm[r][w0e[e]]*omf + rowm[r][w1e[e]]*fwe[e];
      float su = rows[r][w0e[e]]*omf + rows[r][w1e[e]]*fwe[e];
      float enh = (xc - mu) / (su + 1e-6f) * a + bb;
      res[e] = clip10(xc + 0.1f*enh);
    }
    float4 o = {res[0], res[1], res[2], res[3]};
    *(float4*)(&orow[(size_t)r*W_ + tid*4]) = o;   // coalesced b128 store
  }
}

extern "C" void kernel_launch(void* const* d_in, const int* in_sizes, int n_in,
                              void* d_out, int out_size, void* d_ws, size_t ws_size,
                              hipStream_t stream) {
  (void)in_sizes; (void)n_in; (void)out_size; (void)ws_size;
  const float* x     = (const float*)d_in[0];
  const float* alpha = (const float*)d_in[1];
  const float* beta  = (const float*)d_in[2];
  float* out  = (float*)d_out;
  float* mean = (float*)d_ws;              // 8192 f32 (32KB)
  float* stdt = mean + NPLANE*64;          // 8192 f32 (32KB)

  cell_mean_kernel<<<NPLANE*64, 320, 0, stream>>>(x, mean);
  cell_std_kernel <<<NPLANE*64, 320, 0, stream>>>(x, mean, stdt);
  apply_kernel    <<<NPLANE*(H_/4), 160, 0, stream>>>(x, mean, stdt,
                                                      alpha, beta, out);
}